// MultiScaleBSQ_4389456577200
// MI455X (gfx1250) — compile-verified
//
#include <hip/hip_runtime.h>
#include <math.h>

// ---------------- problem constants ----------------
#define NB   32          // batch
#define NT   8192        // time
#define NC   32          // channels (== wave32!)
#define NLEV 14
#define TOTROWS 16383L   // sum of SCALE_SCHEDULE = 2^14 - 1
#define QSC  0.17677669529663687f   // 1/sqrt(32)
#define EPSL 1e-8f

typedef float v2f __attribute__((ext_vector_type(2)));
typedef float v8f __attribute__((ext_vector_type(8)));

// ---------------- helpers ----------------
__device__ __forceinline__ float wave_sum(float v) {
#pragma unroll
  for (int m = 16; m >= 1; m >>= 1) v += __shfl_xor(v, m, 32);
  return v;
}

// ---------------- init: residual = f, zero accumulators ----------------
__global__ void msbsq_init_kernel(const float* __restrict__ f,
                                  float* __restrict__ resid,
                                  float* __restrict__ acc) {
  long i = (long)blockIdx.x * blockDim.x + threadIdx.x;
  const long n4 = (long)NB * NT * NC / 4;
  if (i < n4) ((float4*)resid)[i] = ((const float4*)f)[i];
  if (i < NLEV * 64) acc[i] = 0.0f;   // [ent, commit, pad.., p[32]] per level
}

__global__ void msbsq_zero_kernel(float* __restrict__ p, long n) {
  long i = (long)blockIdx.x * blockDim.x + threadIdx.x;
  if (i < n) p[i] = 0.0f;
}

// ---------------- WMMA area pooling ----------------
// One wave pools `chunk` time steps of one token's window using
// V_WMMA_F32_16X16X4_F32 with A = (1/W) * ones  (ones-matmul column reduction).
// B-tile native layout: lane n<16 -> K={0,1} chan n ; lane n+16 -> K={2,3} chan n.
__global__ void msbsq_pool_wmma_kernel(const float* __restrict__ resid,
                                       float* __restrict__ r,
                                       int pt, int nSub, int chunk) {
  const int lane = threadIdx.x & 31;
  const int wid  = (int)((blockIdx.x * blockDim.x + threadIdx.x) >> 5);
  const int nTok = NB * pt;
  const bool active = wid < nTok * nSub;

  const int W = NT / pt;
  int tok = 0, sub = 0;
  if (active) { tok = wid / nSub; sub = wid - tok * nSub; }
  const int b = tok / pt, j = tok - b * pt;
  const int t0 = j * W + sub * chunk;

  // per-lane base: time row = 2*(lane>>4), channel = lane&15
  const int rowSel = (lane >> 4) << 1;
  const int col = lane & 15;
  const float* base = resid + (((long)b * NT + t0 + rowSel) * NC) + col;

  const float invW = 1.0f / (float)W;
  v2f a; a[0] = invW; a[1] = invW;        // A tile: every element 1/W
  v8f accLo = {}; v8f accHi = {};

  if (active) {
    const int iters = chunk >> 2;         // 4 time steps per WMMA
    for (int i = 0; i < iters; ++i) {
      const float* q = base + (long)i * 4 * NC;
      v2f bLo, bHi;
      bLo[0] = q[0];        // (t0+rowSel  , chan col)
      bLo[1] = q[NC];       // (t0+rowSel+1, chan col)
      bHi[0] = q[16];       // (t0+rowSel  , chan col+16)
      bHi[1] = q[NC + 16];  // (t0+rowSel+1, chan col+16)
      accLo = __builtin_amdgcn_wmma_f32_16x16x4_f32(false, a, false, bLo,
                                                    (short)0, accLo, false, false);
      accHi = __builtin_amdgcn_wmma_f32_16x16x4_f32(false, a, false, bHi,
                                                    (short)0, accHi, false, false);
    }
    // D rows are identical column sums. VGPR0: lanes0-15 = chan 0..15,
    // accHi VGPR0 lanes16-31 = chan 16..31 -> lane L holds channel L.
    float val = (lane < 16) ? accLo[0] : accHi[0];
    float* dst = r + (long)tok * NC + lane;
    if (nSub == 1) *dst = val;
    else atomicAdd(dst, val);
  }
}

// ---------------- BSQ per-token math (wave per token, lane = channel) -------
__global__ void msbsq_bsq_kernel(const float* __restrict__ src, int pooled,
                                 int pt,
                                 float* __restrict__ acc,       // level accumulators
                                 int* __restrict__ bitsOut) {
  __shared__ float lds_p[NC];
  __shared__ float lds_s[2];
  const int tid = threadIdx.x, lane = tid & 31;
  if (tid < NC) lds_p[tid] = 0.0f;
  if (tid < 2)  lds_s[tid] = 0.0f;
  __syncthreads();

  const int wid = (int)((blockIdx.x * blockDim.x + tid) >> 5);
  const int nTok = NB * pt;
  const bool active = wid < nTok;

  float z = 0.0f;
  int b = 0, j = 0;
  if (active) {
    b = wid / pt; j = wid - b * pt;
    if (pooled) {
      z = src[(long)wid * NC + lane];
    } else {
      const int W = NT / pt;                 // 1 or 2 here
      const long base = ((long)b * NT + (long)j * W) * NC + lane;
      z = (W == 1) ? src[base] : 0.5f * (src[base] + src[base + NC]);
    }
  }

  // normalize over C (cross-lane, wave32)
  float n2 = wave_sum(z * z);
  float inv = 1.0f / fmaxf(sqrtf(n2), 1e-12f);
  float zn = z * inv;

  const int bit = (zn > 0.0f) ? 1 : 0;
  const float zhat = bit ? QSC : -QSC;
  const float d = zhat - zn;
  float commit = d * d;

  // p = sigmoid(-4 * z / sqrt(C) * INV_TEMP) = 1/(1+exp(400*QSC*zn))
  const float x = 400.0f * QSC * zn;
  const float p = 1.0f / (1.0f + __expf(x));
  const float op = 1.0f - p;
  float e = -(p * __logf(p + EPSL)) - (op * __logf(op + EPSL));

  float es = wave_sum(e);
  float cs = wave_sum(commit);

  if (active) {
    bitsOut[((long)b * TOTROWS + (pt - 1) + j) * NC + lane] = bit;
    atomicAdd(&lds_p[lane], p);
    if (lane == 0) { atomicAdd(&lds_s[0], es); atomicAdd(&lds_s[1], cs); }
  }
  __syncthreads();
  if (tid < NC) atomicAdd(&acc[32 + tid], lds_p[tid]);
  if (tid == 0) { atomicAdd(&acc[0], lds_s[0]); atomicAdd(&acc[1], lds_s[1]); }
}

// ---------------- residual -= linear_up(quant) (quant from sign bits) -------
__global__ void msbsq_update_kernel(float* __restrict__ resid,
                                    const int* __restrict__ bits,
                                    int pt) {
  long idx = (long)blockIdx.x * blockDim.x + threadIdx.x; // (b, t, chan-quad)
  const long total = (long)NB * NT * (NC / 4);
  if (idx >= total) return;
  const int cq = (int)(idx & 7);
  const long bt = idx >> 3;
  const int t = (int)(bt & (NT - 1));
  const int b = (int)(bt >> 13);

  const float invW = (float)pt / (float)NT;
  float pos = ((float)t + 0.5f) * invW - 0.5f;
  pos = fminf(fmaxf(pos, 0.0f), (float)(pt - 1));
  const int lo = (int)pos;
  const float w = pos - (float)lo;
  const int hi = min(lo + 1, pt - 1);
  const long off = (long)(pt - 1);

  const int4 l = ((const int4*)(bits + ((long)b * TOTROWS + off + lo) * NC))[cq];
  const int4 h = ((const int4*)(bits + ((long)b * TOTROWS + off + hi) * NC))[cq];
  float4* rp = (float4*)(resid + bt * NC) + cq;
  float4 rv = *rp;

  float ql, qh;
  ql = l.x ? QSC : -QSC; qh = h.x ? QSC : -QSC; rv.x -= ql + w * (qh - ql);
  ql = l.y ? QSC : -QSC; qh = h.y ? QSC : -QSC; rv.y -= ql + w * (qh - ql);
  ql = l.z ? QSC : -QSC; qh = h.z ? QSC : -QSC; rv.z -= ql + w * (qh - ql);
  ql = l.w ? QSC : -QSC; qh = h.w ? QSC : -QSC; rv.w -= ql + w * (qh - ql);
  *rp = rv;
}

// ---------------- quantized_out = f - residual_final ----------------
__global__ void msbsq_final_kernel(const float* __restrict__ f,
                                   const float* __restrict__ resid,
                                   float* __restrict__ qout) {
  long i = (long)blockIdx.x * blockDim.x + threadIdx.x;
  const long n4 = (long)NB * NT * NC / 4;
  if (i >= n4) return;
  float4 a = ((const float4*)f)[i];
  float4 r = ((const float4*)resid)[i];
  float4 o = {a.x - r.x, a.y - r.y, a.z - r.z, a.w - r.w};
  ((float4*)qout)[i] = o;
}

// ---------------- per-level loss finalize (one wave per level) --------------
__global__ void msbsq_loss_kernel(const float* __restrict__ acc,
                                  float* __restrict__ lossOut) {
  const int L = blockIdx.x;
  const int lane = threadIdx.x & 31;
  const float* a = acc + L * 64;
  const int pt = 1 << L;
  const float N = (float)(NB * pt);
  const float ap = a[32 + lane] / N;
  const float o = 1.0f - ap;
  float ce = -(ap * __logf(ap + EPSL)) - (o * __logf(o + EPSL));
  ce = wave_sum(ce);
  if (lane == 0) {
    const float per_sample = a[0] / N;
    const float commit = a[1] / N;
    const float penalty = (per_sample - ce) / 100.0f;   // / INV_TEMP
    lossOut[L] = penalty * 0.1f + commit * 0.2f;        // ENTROPY_W, COMMIT_W
  }
}

// ---------------- host orchestration ----------------
extern "C" void kernel_launch(void* const* d_in, const int* in_sizes, int n_in,
                              void* d_out, int out_size, void* d_ws, size_t ws_size,
                              hipStream_t stream) {
  (void)in_sizes; (void)n_in; (void)out_size; (void)ws_size;
  const float* f = (const float*)d_in[0];
  float* out = (float*)d_out;

  const long QN = (long)NB * NT * NC;               // 8,388,608
  const long BITSN = (long)NB * TOTROWS * NC;       // 16,776,192
  int*   bitsOut = (int*)(out + QN);
  float* lossOut = out + QN + BITSN;

  float* acc   = (float*)d_ws;                          // 14*64 floats
  float* resid = (float*)((char*)d_ws + 4096);          // 33.5 MB
  float* rbuf  = (float*)((char*)d_ws + 4096 + QN * 4); // <= 8 MB pooled buf

  const int blk = 256;
  { long g = (QN / 4 + blk - 1) / blk;
    msbsq_init_kernel<<<(unsigned)g, blk, 0, stream>>>(f, resid, acc); }

  for (int L = 0; L < NLEV; ++L) {
    const int pt = 1 << L;
    const int W = NT >> L;
    const float* src; int pooled;
    if (W >= 4) {
      const int chunk = (W > 256) ? 256 : W;
      const int nSub = W / chunk;
      if (nSub > 1) {
        long n = (long)NB * pt * NC;
        msbsq_zero_kernel<<<(unsigned)((n + blk - 1) / blk), blk, 0, stream>>>(rbuf, n);
      }
      long thr = (long)NB * pt * nSub * 32;
      msbsq_pool_wmma_kernel<<<(unsigned)((thr + blk - 1) / blk), blk, 0, stream>>>(
          resid, rbuf, pt, nSub, chunk);
      src = rbuf; pooled = 1;
    } else {
      src = resid; pooled = 0;
    }
    { long thr = (long)NB * pt * 32;
      msbsq_bsq_kernel<<<(unsigned)((thr + blk - 1) / blk), blk, 0, stream>>>(
          src, pooled, pt, acc + L * 64, bitsOut); }
    { long thr = (long)NB * NT * (NC / 4);
      msbsq_update_kernel<<<(unsigned)((thr + blk - 1) / blk), blk, 0, stream>>>(
          resid, bitsOut, pt); }
  }

  { long g = (QN / 4 + blk - 1) / blk;
    msbsq_final_kernel<<<(unsigned)g, blk, 0, stream>>>(f, resid, out); }
  msbsq_loss_kernel<<<NLEV, 32, 0, stream>>>(acc, lossOut);
}